// Model_36249523978795
// MI455X (gfx1250) — compile-verified
//
#include <hip/hip_runtime.h>
#include <math.h>

typedef __attribute__((ext_vector_type(2))) float v2f;
typedef __attribute__((ext_vector_type(8))) float v8f;

#define N_NODES 1024
#define HDIM 64
#define MDIM 32
#define NSPLIT 4
#define JCHUNK 128

__device__ __forceinline__ v8f wmma_f32_16x16x4(v2f a, v2f b, v8f c) {
  // D = A(16x4 f32) x B(4x16 f32) + C(16x16 f32), exact f32 on CDNA5
  return __builtin_amdgcn_wmma_f32_16x16x4_f32(false, a, false, b, (short)0, c,
                                               false, false);
}

// ---------------------------------------------------------------------------
// h = features + sinusoidal positional encoding
// ---------------------------------------------------------------------------
__global__ void posenc_add_kernel(const float* __restrict__ feat,
                                  float* __restrict__ h) {
  int idx = blockIdx.x * blockDim.x + threadIdx.x;
  if (idx >= N_NODES * HDIM) return;
  int n = idx >> 6;
  int d = idx & 63;
  int tp = d >> 1;
  float div = expf((float)(2 * tp) * (-logf(10000.0f) / (float)HDIM));
  float ang = (float)n * div;
  float pe = (d & 1) ? cosf(ang) : sinf(ang);
  h[idx] = feat[idx] + pe;
}

// ---------------------------------------------------------------------------
// A = h @ Wa.T ; Bm = h @ Wb.T  where Wa = w_msg[:, :64], Wb = w_msg[:, 64:128]
// One wave per 16x16 output tile, K=64 in 16 steps of V_WMMA_F32_16X16X4_F32.
// wave 0/1 -> A cols 0-15 / 16-31 ; wave 2/3 -> Bm cols 0-15 / 16-31
// ---------------------------------------------------------------------------
__global__ void proj_kernel(const float* __restrict__ h,
                            const float* __restrict__ wmsg,
                            float* __restrict__ Aout,
                            float* __restrict__ Bout) {
  const int lane = threadIdx.x & 31;
  const int wave = threadIdx.x >> 5;   // 0..3
  const int itile = blockIdx.x;        // 0..63
  const int l16 = lane & 15;
  const int khalf = (lane >> 4) << 1;  // 0 (lanes 0-15) or 2 (lanes 16-31)
  const int kbase = (wave & 1) << 4;
  const int woff = (wave >> 1) << 6;   // 0 for Wa, 64 for Wb
  const float* hrow = h + (itile * 16 + l16) * HDIM;
  const float* wrow = wmsg + (kbase + l16) * (2 * HDIM + 1) + woff;
  v8f c = {};
#pragma unroll
  for (int d = 0; d < HDIM; d += 4) {
    v2f a, b;
    a.x = hrow[d + khalf];       // A: lane = M row, vgpr0/1 = K, K+1 (halves +2)
    a.y = hrow[d + khalf + 1];
    b.x = wrow[d + khalf];       // B: lane = N col, vgpr0/1 = K row, K+1
    b.y = wrow[d + khalf + 1];
    c = wmma_f32_16x16x4(a, b, c);
  }
  float* out = (wave >> 1) ? Bout : Aout;
  const int col = kbase + l16;
  const int rbase = itile * 16 + ((lane >> 4) << 3);
#pragma unroll
  for (int vr = 0; vr < 8; vr++)
    out[(rbase + vr) * MDIM + col] = c[vr];
}

// ---------------------------------------------------------------------------
// Pairwise message pass. Block = 256 threads = one 16-row i-tile, one of
// NSPLIT j ranges. Thread (i0,k) owns rows i0 and i0+8 for channel k and
// accumulates msum[i,k] and T[i,k,c] = sum_j m*x_j[c] in registers.
// ---------------------------------------------------------------------------
__global__ void pair_kernel(const float* __restrict__ x,
                            const float* __restrict__ Aproj,
                            const float* __restrict__ Bproj,
                            const float* __restrict__ wmsg,
                            const float* __restrict__ bmsg,
                            const unsigned char* __restrict__ mask,
                            float* __restrict__ msum_part,
                            float* __restrict__ T_part) {
  __shared__ float sA[JCHUNK * MDIM];
  __shared__ float sx[JCHUNK * 3];
  __shared__ float sd2[JCHUNK * 16];
  __shared__ float smj[JCHUNK];
  __shared__ float sxi[16 * 3];
  __shared__ float smi[16];
  const int t = threadIdx.x;  // 256
  const int k = t & 31;
  const int i0 = t >> 5;      // 0..7
  const int itile = blockIdx.x;
  const int split = blockIdx.y;
  const int jsplit0 = split * (N_NODES / NSPLIT);

  if (t < 16) {
    int gi = itile * 16 + t;
    sxi[t * 3 + 0] = x[gi * 3 + 0];
    sxi[t * 3 + 1] = x[gi * 3 + 1];
    sxi[t * 3 + 2] = x[gi * 3 + 2];
    smi[t] = mask[gi] ? 1.0f : 0.0f;
  }
  __syncthreads();

  const int ia = itile * 16 + i0;
  const int ib = ia + 8;
  const float wdk = wmsg[k * (2 * HDIM + 1) + 2 * HDIM];
  const float bk = bmsg[k];
  const float Ba = Bproj[ia * MDIM + k] + bk;
  const float Bb = Bproj[ib * MDIM + k] + bk;
  const float mia = smi[i0];
  const float mib = smi[i0 + 8];

  float macc_a = 0.f, T0a = 0.f, T1a = 0.f, T2a = 0.f;
  float macc_b = 0.f, T0b = 0.f, T1b = 0.f, T2b = 0.f;

  for (int jc = 0; jc < N_NODES / NSPLIT; jc += JCHUNK) {
    const int jbase = jsplit0 + jc;
    __syncthreads();  // previous chunk fully consumed
    {
      const float4* src = (const float4*)(Aproj + jbase * MDIM);
      float4* dst = (float4*)sA;
      for (int p = t; p < JCHUNK * MDIM / 4; p += 256) dst[p] = src[p];
    }
    for (int p = t; p < JCHUNK * 3; p += 256) sx[p] = x[jbase * 3 + p];
    for (int p = t; p < JCHUNK; p += 256)
      smj[p] = mask[jbase + p] ? 1.0f : 0.0f;
    __syncthreads();
    // cooperative d2 tile: JCHUNK x 16
    for (int p = t; p < JCHUNK * 16; p += 256) {
      int jj = p >> 4, ii = p & 15;
      float dx0 = sx[jj * 3 + 0] - sxi[ii * 3 + 0];
      float dx1 = sx[jj * 3 + 1] - sxi[ii * 3 + 1];
      float dx2 = sx[jj * 3 + 2] - sxi[ii * 3 + 2];
      sd2[jj * 16 + ii] = dx0 * dx0 + dx1 * dx1 + dx2 * dx2;
    }
    __syncthreads();
    for (int j = 0; j < JCHUNK; j++) {
      const float xj0 = sx[j * 3 + 0];  // broadcast reads
      const float xj1 = sx[j * 3 + 1];
      const float xj2 = sx[j * 3 + 2];
      const float mj = smj[j];
      const int gj = jbase + j;
      const float aA = sA[j * MDIM + k];  // bank = lane, conflict-free
      const float d2a = sd2[j * 16 + i0];
      const float d2b = sd2[j * 16 + i0 + 8];
      const float fa = (gj == ia) ? 0.f : mia * mj;
      const float fb = (gj == ib) ? 0.f : mib * mj;
      float ma = fmaxf(fmaf(d2a, wdk, aA + Ba), 0.f) * fa;
      float mb = fmaxf(fmaf(d2b, wdk, aA + Bb), 0.f) * fb;
      macc_a += ma;
      T0a = fmaf(ma, xj0, T0a);
      T1a = fmaf(ma, xj1, T1a);
      T2a = fmaf(ma, xj2, T2a);
      macc_b += mb;
      T0b = fmaf(mb, xj0, T0b);
      T1b = fmaf(mb, xj1, T1b);
      T2b = fmaf(mb, xj2, T2b);
    }
  }

  msum_part[(split * N_NODES + ia) * MDIM + k] = macc_a;
  msum_part[(split * N_NODES + ib) * MDIM + k] = macc_b;
  float* Tp = T_part + (size_t)((split * N_NODES + ia) * MDIM + k) * 3;
  Tp[0] = T0a; Tp[1] = T1a; Tp[2] = T2a;
  Tp = T_part + (size_t)((split * N_NODES + ib) * MDIM + k) * 3;
  Tp[0] = T0b; Tp[1] = T1b; Tp[2] = T2b;
}

// ---------------------------------------------------------------------------
// Reduce split partials, apply pos head:
// x_out = x + (sum_k wpos[c,k]*(T - x*msum) + bpos[c]*R[c]) / (N-1)
// Also writes total msum for the feature GEMM.
// ---------------------------------------------------------------------------
__global__ void finish_kernel(const float* __restrict__ x_in,
                              const float* __restrict__ wpos,
                              const float* __restrict__ bpos,
                              const unsigned char* __restrict__ mask,
                              const float* __restrict__ msum_part,
                              const float* __restrict__ T_part,
                              float* __restrict__ x_out,
                              float* __restrict__ msum_tot) {
  const int i = blockIdx.x * blockDim.x + threadIdx.x;
  if (i >= N_NODES) return;
  const float xi0 = x_in[i * 3 + 0];
  const float xi1 = x_in[i * 3 + 1];
  const float xi2 = x_in[i * 3 + 2];
  const float mi = mask[i] ? 1.0f : 0.0f;
  float R0 = 0.f, R1 = 0.f, R2 = 0.f;
  for (int j = 0; j < N_NODES; j++) {
    float f = (j == i) ? 0.f : (mask[j] ? mi : 0.f);
    R0 += (x_in[j * 3 + 0] - xi0) * f;
    R1 += (x_in[j * 3 + 1] - xi1) * f;
    R2 += (x_in[j * 3 + 2] - xi2) * f;
  }
  float acc0 = 0.f, acc1 = 0.f, acc2 = 0.f;
  for (int k = 0; k < MDIM; k++) {
    float ms = 0.f, T0 = 0.f, T1 = 0.f, T2 = 0.f;
    for (int s = 0; s < NSPLIT; s++) {
      const int base = (s * N_NODES + i) * MDIM + k;
      ms += msum_part[base];
      const float* Tp = T_part + (size_t)base * 3;
      T0 += Tp[0]; T1 += Tp[1]; T2 += Tp[2];
    }
    msum_tot[i * MDIM + k] = ms;
    acc0 = fmaf(wpos[0 * MDIM + k], T0 - xi0 * ms, acc0);
    acc1 = fmaf(wpos[1 * MDIM + k], T1 - xi1 * ms, acc1);
    acc2 = fmaf(wpos[2 * MDIM + k], T2 - xi2 * ms, acc2);
  }
  const float inv = 1.0f / (float)(N_NODES - 1);
  x_out[i * 3 + 0] = xi0 + (acc0 + bpos[0] * R0) * inv;
  x_out[i * 3 + 1] = xi1 + (acc1 + bpos[1] * R1) * inv;
  x_out[i * 3 + 2] = xi2 + (acc2 + bpos[2] * R2) * inv;
}

// ---------------------------------------------------------------------------
// h1 = relu([h | msum] @ w_feat.T + b_feat)   (layer 1 only; layer 2's h dies)
// WMMA f32 GEMM, K = 96 in 24 steps.
// ---------------------------------------------------------------------------
__global__ void feat_kernel(const float* __restrict__ h,
                            const float* __restrict__ msum,
                            const float* __restrict__ wfeat,
                            const float* __restrict__ bfeat,
                            float* __restrict__ hout) {
  const int lane = threadIdx.x & 31;
  const int wave = threadIdx.x >> 5;  // o-tile 0..3
  const int itile = blockIdx.x;
  const int l16 = lane & 15;
  const int khalf = (lane >> 4) << 1;
  const int row = itile * 16 + l16;
  const float* hrow = h + row * HDIM;
  const float* mrow = msum + row * MDIM;
  const int ocol = wave * 16 + l16;
  const float* wrow = wfeat + ocol * (HDIM + MDIM);
  v8f c = {};
#pragma unroll
  for (int d = 0; d < HDIM + MDIM; d += 4) {
    const int dd = d + khalf;  // dd, dd+1 never straddle the h|msum boundary
    v2f a, b;
    a.x = (dd < HDIM) ? hrow[dd] : mrow[dd - HDIM];
    a.y = (dd + 1 < HDIM) ? hrow[dd + 1] : mrow[dd + 1 - HDIM];
    b.x = wrow[dd];
    b.y = wrow[dd + 1];
    c = wmma_f32_16x16x4(a, b, c);
  }
  const float bias = bfeat[ocol];
  const int rbase = itile * 16 + ((lane >> 4) << 3);
#pragma unroll
  for (int vr = 0; vr < 8; vr++) {
    float v = c[vr] + bias;
    hout[(rbase + vr) * HDIM + ocol] = fmaxf(v, 0.0f);
  }
}

// ---------------------------------------------------------------------------
extern "C" void kernel_launch(void* const* d_in, const int* in_sizes, int n_in,
                              void* d_out, int out_size, void* d_ws,
                              size_t ws_size, hipStream_t stream) {
  (void)in_sizes; (void)n_in; (void)out_size; (void)ws_size;
  const float* positions = (const float*)d_in[0];
  const float* features = (const float*)d_in[1];
  const unsigned char* mask = (const unsigned char*)d_in[2];
  const float* w_msg1 = (const float*)d_in[3];
  const float* b_msg1 = (const float*)d_in[4];
  const float* w_pos1 = (const float*)d_in[5];
  const float* b_pos1 = (const float*)d_in[6];
  const float* w_feat1 = (const float*)d_in[7];
  const float* b_feat1 = (const float*)d_in[8];
  const float* w_msg2 = (const float*)d_in[9];
  const float* b_msg2 = (const float*)d_in[10];
  const float* w_pos2 = (const float*)d_in[11];
  const float* b_pos2 = (const float*)d_in[12];
  // w_feat2 / b_feat2 / t unused: layer-2 h is dead code (only x returned)

  float* ws = (float*)d_ws;
  float* h0 = ws;        ws += N_NODES * HDIM;
  float* h1 = ws;        ws += N_NODES * HDIM;
  float* Aproj = ws;     ws += N_NODES * MDIM;
  float* Bproj = ws;     ws += N_NODES * MDIM;
  float* msum_part = ws; ws += NSPLIT * N_NODES * MDIM;
  float* T_part = ws;    ws += NSPLIT * N_NODES * MDIM * 3;
  float* msum_tot = ws;  ws += N_NODES * MDIM;
  float* x1 = ws;        ws += N_NODES * 3;
  float* x_out = (float*)d_out;

  posenc_add_kernel<<<(N_NODES * HDIM + 255) / 256, 256, 0, stream>>>(features, h0);

  // ---- layer 1 ----
  proj_kernel<<<64, 128, 0, stream>>>(h0, w_msg1, Aproj, Bproj);
  pair_kernel<<<dim3(64, NSPLIT), 256, 0, stream>>>(positions, Aproj, Bproj,
                                                    w_msg1, b_msg1, mask,
                                                    msum_part, T_part);
  finish_kernel<<<(N_NODES + 255) / 256, 256, 0, stream>>>(
      positions, w_pos1, b_pos1, mask, msum_part, T_part, x1, msum_tot);
  feat_kernel<<<64, 128, 0, stream>>>(h0, msum_tot, w_feat1, b_feat1, h1);

  // ---- layer 2 (only x is needed) ----
  proj_kernel<<<64, 128, 0, stream>>>(h1, w_msg2, Aproj, Bproj);
  pair_kernel<<<dim3(64, NSPLIT), 256, 0, stream>>>(x1, Aproj, Bproj, w_msg2,
                                                    b_msg2, mask, msum_part,
                                                    T_part);
  finish_kernel<<<(N_NODES + 255) / 256, 256, 0, stream>>>(
      x1, w_pos2, b_pos2, mask, msum_part, T_part, x_out, msum_tot);
}